// DTEMDNet_49692771614875
// MI455X (gfx1250) — compile-verified
//
#include <hip/hip_runtime.h>

// MI455X / gfx1250, wave32. D = A(16x4 f32) x B(4x16 f32) + C(16x16 f32),
// eight chained WMMAs cover K=32. Kernel is output-store bound:
//   write 262144*900*4 B = 943 MB @ 23.3 TB/s  ->  ~42 us floor.

typedef float v2f __attribute__((ext_vector_type(2)));
typedef float v8f __attribute__((ext_vector_type(8)));

#define KDIM   32
#define NCOLS  900     // S*S, S = 30
#define S_IMG  30
#define NTILES 57      // ceil(900/16)

__global__ __launch_bounds__(256)
void dtemd_wmma_serp(const float* __restrict__ A,     // sparse_code [B,32]
                     const float* __restrict__ Dict,  // dictionary  [900,32]
                     float* __restrict__ Out)         // output      [B,900] (serpentine cols)
{
    const int tid   = threadIdx.x;
    const int lane  = tid & 31;
    const int wave  = tid >> 5;          // 8 waves / block
    const int lcol  = lane & 15;         // N (and M) index within tile
    const int lhalf = lane >> 4;         // 0: K={4j,4j+1}, 1: K={4j+2,4j+3}

    // Each wave owns 16 batch rows.
    const long rowBase = ((long)blockIdx.x * 8 + wave) * 16;

    // ---- Load A fragments: 8 K-slices of the 16x32 row block ----
    // lane L (and L+16) hold row M = L&15; each lane keeps 16 of the 32 K values.
    v2f a[8];
    const float* arow = A + (rowBase + lcol) * KDIM + 2 * lhalf;
#pragma unroll
    for (int j = 0; j < 8; ++j)
        a[j] = *(const v2f*)(arow + 4 * j);

    // ---- Sweep all 57 dictionary tiles ----
    for (int t = 0; t < NTILES; ++t) {
        const int nBase = t * 16;
        const int n     = nBase + lcol;          // dictionary atom / output column
        const int nc    = (n < NCOLS) ? n : (NCOLS - 1); // clamp OOB lanes (stores masked)

        // B fragment: B[k][n] = Dict[n][k]; K contiguous in memory -> b64 loads.
        // Dict is 115 KB: permanently L2-resident, so these hit L2.
        v2f b[8];
        const float* brow = Dict + (long)nc * KDIM + 2 * lhalf;
#pragma unroll
        for (int j = 0; j < 8; ++j)
            b[j] = *(const v2f*)(brow + 4 * j);

        // ---- 8 x V_WMMA_F32_16X16X4_F32, accumulate K = 32 ----
        // EXEC is all-1s here (uniform control flow), as WMMA requires.
        v8f acc = {};
#pragma unroll
        for (int j = 0; j < 8; ++j)
            acc = __builtin_amdgcn_wmma_f32_16x16x4_f32(
                      /*neg_a=*/false, a[j],
                      /*neg_b=*/false, b[j],
                      /*c_mod=*/(short)0, acc,
                      /*reuse_a=*/false, /*reuse_b=*/false);

        // ---- Serpentine store: out[b, r, (r odd ? 29-c : c)] = samples[b, n] ----
        // C/D layout: VGPR v holds (M = v + 8*lhalf, N = lcol). Column permutation
        // is within a 30-wide row, but tiles can straddle two rows, so lanes 0..15 cover one or two
        // contiguous runs per store (each ascending or reversed). Nontemporal: 943 MB stream,
        // never re-read by this kernel -> keep it out of L2's working set.
        if (n < NCOLS) {
            const int r  = n / S_IMG;
            const int c  = n - r * S_IMG;
            const int pn = (r & 1) ? (r * S_IMG + (S_IMG - 1) - c) : n;
            float* outp = Out + (rowBase + 8 * lhalf) * (long)NCOLS + pn;
#pragma unroll
            for (int v = 0; v < 8; ++v)
                __builtin_nontemporal_store(acc[v], outp + (long)v * NCOLS);
        }
    }
}

extern "C" void kernel_launch(void* const* d_in, const int* in_sizes, int n_in,
                              void* d_out, int out_size, void* d_ws, size_t ws_size,
                              hipStream_t stream) {
    const float* sparse = (const float*)d_in[0];   // [B, 32] fp32
    const float* dict   = (const float*)d_in[1];   // [900, 32] fp32
    // d_in[2] = image_size scalar (30), baked in as a compile-time constant.
    float* out = (float*)d_out;                    // [B, 1, 30, 30] fp32

    const int B = in_sizes[0] / KDIM;              // 262144
    const int blocks = B / (16 * 8);               // 16 rows/wave * 8 waves/block = 2048 blocks

    hipLaunchKernelGGL(dtemd_wmma_serp, dim3(blocks), dim3(256), 0, stream,
                       sparse, dict, out);
}